// GNNModel_13262859010050
// MI455X (gfx1250) — compile-verified
//
#include <hip/hip_runtime.h>
#include <math.h>

#define NN1 100000
#define NN2 100000
#define NE  3200000

typedef __attribute__((ext_vector_type(2))) float v2f;
typedef __attribute__((ext_vector_type(8))) float v8f;

// ---------------- monotonic float<->uint key for atomic segment-max ----------
__device__ __forceinline__ unsigned fmax_key(float x) {
  unsigned u = __float_as_uint(x);
  return (u & 0x80000000u) ? ~u : (u | 0x80000000u);
}
__device__ __forceinline__ float fmax_unkey(unsigned k) {
  return __uint_as_float((k & 0x80000000u) ? (k & 0x7FFFFFFFu) : ~k);
}

// ---------------- layer-1 linear (in_dim = 1): two outputs per pass ----------
__global__ __launch_bounds__(256) void lin1_dual(
    const float* __restrict__ x, int n,
    const float* __restrict__ Wa, const float* __restrict__ ba, float* __restrict__ outa,
    const float* __restrict__ Wb, const float* __restrict__ bb, float* __restrict__ outb)
{
  int idx = blockIdx.x * blockDim.x + threadIdx.x;
  if (idx >= n * 16) return;
  int i = idx >> 4, f = idx & 15;
  float xv = x[i];
  outa[idx] = fmaf(xv, Wa[f], ba[f]);
  outb[idx] = fmaf(xv, Wb[f], bb[f]);
}

// ---------------- WMMA node transform: x[N,16] @ W[16,16] + b, two GEMMs -----
// One wave handles a 16-node tile; A (the x tile) is shared by both GEMMs.
// 8x v_wmma_f32_16x16x4_f32 per tile.
__global__ __launch_bounds__(128) void node_xform16_dual(
    const float* __restrict__ x, int ntiles,
    const float* __restrict__ Wa, const float* __restrict__ ba, float* __restrict__ outa,
    const float* __restrict__ Wb, const float* __restrict__ bb, float* __restrict__ outb)
{
  int lane = threadIdx.x & 31;
  int wave = threadIdx.x >> 5;
  int tile = blockIdx.x * 4 + wave;
  if (tile >= ntiles) return;          // whole-wave uniform exit: EXEC stays all-ones
  int col  = lane & 15;                // N (for B/C/D) and M (for A)
  int half = lane >> 4;                // selects K {0,1} vs {2,3} within a K=4 slice

  const float* xrow = x + (size_t)(tile * 16 + col) * 16;  // A: lane holds row M=col
  float bav = ba[col], bbv = bb[col];
  v8f acca, accb;
#pragma unroll
  for (int j = 0; j < 8; ++j) { acca[j] = bav; accb[j] = bbv; }

#pragma unroll
  for (int kk = 0; kk < 4; ++kk) {
    int k0 = kk * 4 + half * 2;
    v2f A;  A.x  = xrow[k0];              A.y  = xrow[k0 + 1];
    v2f Ba; Ba.x = Wa[k0 * 16 + col];     Ba.y = Wa[(k0 + 1) * 16 + col];
    v2f Bb; Bb.x = Wb[k0 * 16 + col];     Bb.y = Wb[(k0 + 1) * 16 + col];
    acca = __builtin_amdgcn_wmma_f32_16x16x4_f32(false, A, false, Ba, (short)0, acca, false, false);
    accb = __builtin_amdgcn_wmma_f32_16x16x4_f32(false, A, false, Bb, (short)0, accb, false, false);
  }

#pragma unroll
  for (int j = 0; j < 8; ++j) {
    int m = j + half * 8;
    outa[(size_t)(tile * 16 + m) * 16 + col] = acca[j];
    outb[(size_t)(tile * 16 + m) * 16 + col] = accb[j];
  }
}

// ---------------- per-dst segment state init --------------------------------
__global__ __launch_bounds__(256) void seg_init(unsigned* __restrict__ mkey,
                                                float* __restrict__ s, int n) {
  int i = blockIdx.x * blockDim.x + threadIdx.x;
  if (i < n) { mkey[i] = 0u; s[i] = 0.0f; }
}

// ---------------- edge pass 1: alpha + segment max --------------------------
__global__ __launch_bounds__(256) void edge_alpha(
    const float* __restrict__ q, const float* __restrict__ k,
    const float* __restrict__ We, const float* __restrict__ ea,
    const int* __restrict__ si, const int* __restrict__ di,
    float* __restrict__ alpha, unsigned* __restrict__ mkey, int ne)
{
  int e = blockIdx.x * blockDim.x + threadIdx.x;
  if (e >= ne) return;
  int sn = si[e], dn = di[e];
  float ew = ea[e];
  const float4* qv = (const float4*)(q + (size_t)dn * 16);
  const float4* kv = (const float4*)(k + (size_t)sn * 16);
  const float4* wv = (const float4*)We;
  float acc = 0.0f;
#pragma unroll
  for (int r = 0; r < 4; ++r) {
    float4 a = qv[r], b = kv[r], w = wv[r];
    acc += a.x * fmaf(ew, w.x, b.x) + a.y * fmaf(ew, w.y, b.y)
         + a.z * fmaf(ew, w.z, b.z) + a.w * fmaf(ew, w.w, b.w);
  }
  acc *= 0.25f;                        // 1/sqrt(16)
  alpha[e] = acc;
  atomicMax(&mkey[dn], fmax_key(acc));
}

// ---------------- edge pass 2: exp(alpha - max) + segment sum ---------------
__global__ __launch_bounds__(256) void edge_exp(
    const unsigned* __restrict__ mkey, const int* __restrict__ di,
    float* __restrict__ alpha, float* __restrict__ s, int ne)
{
  int e = blockIdx.x * blockDim.x + threadIdx.x;
  if (e >= ne) return;
  int dn = di[e];
  float a = __expf(alpha[e] - fmax_unkey(mkey[dn]));
  alpha[e] = a;
  atomicAdd(&s[dn], a);
}

// ---------------- edge pass 3: weighted message scatter ---------------------
__global__ __launch_bounds__(256) void edge_scatter(
    const float* __restrict__ v, const float* __restrict__ We,
    const float* __restrict__ ea, const int* __restrict__ si,
    const int* __restrict__ di, const float* __restrict__ alpha,
    const float* __restrict__ s, float* __restrict__ nx, int ne)
{
  int e = blockIdx.x * blockDim.x + threadIdx.x;
  if (e >= ne) return;
  int sn = si[e], dn = di[e];
  float coef = alpha[e] / (s[dn] + 1e-16f);
  float ew = ea[e];
  const float* vr = v + (size_t)sn * 16;
  float* o = nx + (size_t)dn * 16;
#pragma unroll
  for (int c = 0; c < 16; ++c)
    atomicAdd(&o[c], fmaf(ew, We[c], vr[c]) * coef);
}

// ---------------- relu ------------------------------------------------------
__global__ __launch_bounds__(256) void relu_k(float* __restrict__ x, int n) {
  int i = blockIdx.x * blockDim.x + threadIdx.x;
  if (i < n) x[i] = fmaxf(x[i], 0.0f);
}

// ---------------- final fc: [N,16] @ [16,1] + b -----------------------------
__global__ __launch_bounds__(256) void fc_out(
    const float* __restrict__ xh, const float* __restrict__ fw,
    const float* __restrict__ fb, float* __restrict__ out, int n)
{
  int i = blockIdx.x * blockDim.x + threadIdx.x;
  if (i >= n) return;
  const float4* xr = (const float4*)(xh + (size_t)i * 16);
  const float4* wr = (const float4*)fw;
  float acc = fb[0];
#pragma unroll
  for (int r = 0; r < 4; ++r) {
    float4 xv = xr[r], wv = wr[r];
    acc += xv.x * wv.x + xv.y * wv.y + xv.z * wv.z + xv.w * wv.w;
  }
  out[i] = acc;
}

// ---------------- shared edge-pass sequence (one TransformerConv) -----------
static void run_edge_passes(const float* q, const float* k, const float* v,
                            const float* We, const float* ea,
                            const int* si, const int* di,
                            float* alpha, unsigned* mkey, float* s, float* nx,
                            int nd, hipStream_t stream)
{
  const int tb = 256;
  seg_init<<<(nd + tb - 1) / tb, tb, 0, stream>>>(mkey, s, nd);
  edge_alpha<<<(NE + tb - 1) / tb, tb, 0, stream>>>(q, k, We, ea, si, di, alpha, mkey, NE);
  edge_exp<<<(NE + tb - 1) / tb, tb, 0, stream>>>(mkey, di, alpha, s, NE);
  edge_scatter<<<(NE + tb - 1) / tb, tb, 0, stream>>>(v, We, ea, si, di, alpha, s, nx, NE);
  relu_k<<<(nd * 16 + tb - 1) / tb, tb, 0, stream>>>(nx, nd * 16);
}

extern "C" void kernel_launch(void* const* d_in, const int* in_sizes, int n_in,
                              void* d_out, int out_size, void* d_ws, size_t ws_size,
                              hipStream_t stream)
{
  (void)in_sizes; (void)n_in; (void)out_size; (void)ws_size;
  const float* x1  = (const float*)d_in[0];
  const float* x2  = (const float*)d_in[1];
  const float* ea  = (const float*)d_in[2];
  const int*   ei1 = (const int*)d_in[3];
  const int*   ei2 = (const int*)d_in[4];
  const float* W1q = (const float*)d_in[5];
  const float* b1q = (const float*)d_in[6];
  const float* W1k = (const float*)d_in[7];
  const float* b1k = (const float*)d_in[8];
  const float* W1v = (const float*)d_in[9];
  const float* b1v = (const float*)d_in[10];
  const float* W1e = (const float*)d_in[11];
  const float* W1s = (const float*)d_in[12];
  const float* b1s = (const float*)d_in[13];
  const float* W2q = (const float*)d_in[14];
  const float* b2q = (const float*)d_in[15];
  const float* W2k = (const float*)d_in[16];
  const float* b2k = (const float*)d_in[17];
  const float* W2v = (const float*)d_in[18];
  const float* b2v = (const float*)d_in[19];
  const float* W2e = (const float*)d_in[20];
  const float* W2s = (const float*)d_in[21];
  const float* b2s = (const float*)d_in[22];
  const float* fcw = (const float*)d_in[23];
  const float* fcb = (const float*)d_in[24];
  float* out = (float*)d_out;

  // workspace layout (all 64B-aligned slices)
  float* bufA1 = (float*)d_ws;               // [N1,16]
  float* bufA2 = bufA1 + (size_t)NN1 * 16;   // [N2,16]
  float* bufB1 = bufA2 + (size_t)NN2 * 16;
  float* bufB2 = bufB1 + (size_t)NN1 * 16;
  float* qbuf  = bufB2 + (size_t)NN2 * 16;   // [Nmax,16]
  float* kbuf  = qbuf  + (size_t)NN1 * 16;
  float* vbuf  = kbuf  + (size_t)NN1 * 16;
  float* alpha = vbuf  + (size_t)NN1 * 16;   // [E]
  unsigned* mkey = (unsigned*)(alpha + (size_t)NE);  // [Nmax]
  float* sbuf  = (float*)(mkey + NN1);               // [Nmax]

  const int tb = 256;
  const int g1 = (NN1 * 16 + tb - 1) / tb;
  const int g2 = (NN2 * 16 + tb - 1) / tb;
  const int NT1 = NN1 / 16, NT2 = NN2 / 16;  // 100000 % 16 == 0
  const dim3 xg1((NT1 + 3) / 4), xg2((NT2 + 3) / 4), xb(128);

  // ---------- layer 1 (in_dim = 1) ----------
  // conv -> x1 (dst=x1 idx ei1, src=x2 idx ei2), params index 0
  lin1_dual<<<g1, tb, 0, stream>>>(x1, NN1, W1q, b1q, qbuf, W1s, b1s, bufB1);
  lin1_dual<<<g2, tb, 0, stream>>>(x2, NN2, W1k, b1k, kbuf, W1v, b1v, vbuf);
  run_edge_passes(qbuf, kbuf, vbuf, W1e, ea, ei2, ei1, alpha, mkey, sbuf, bufB1, NN1, stream);
  // conv -> x2, params index 1 (offset 16)
  lin1_dual<<<g2, tb, 0, stream>>>(x2, NN2, W1q + 16, b1q + 16, qbuf, W1s + 16, b1s + 16, bufB2);
  lin1_dual<<<g1, tb, 0, stream>>>(x1, NN1, W1k + 16, b1k + 16, kbuf, W1v + 16, b1v + 16, vbuf);
  run_edge_passes(qbuf, kbuf, vbuf, W1e + 16, ea, ei1, ei2, alpha, mkey, sbuf, bufB2, NN2, stream);

  // ---------- layer 2 (in_dim = 16, WMMA transforms), params index 0/1 ----------
  // conv -> x1
  node_xform16_dual<<<xg1, xb, 0, stream>>>(bufB1, NT1, W2q, b2q, qbuf, W2s, b2s, bufA1);
  node_xform16_dual<<<xg2, xb, 0, stream>>>(bufB2, NT2, W2k, b2k, kbuf, W2v, b2v, vbuf);
  run_edge_passes(qbuf, kbuf, vbuf, W2e, ea, ei2, ei1, alpha, mkey, sbuf, bufA1, NN1, stream);
  // conv -> x2, params index 1 (W offset 256, b offset 16)
  node_xform16_dual<<<xg2, xb, 0, stream>>>(bufB2, NT2, W2q + 256, b2q + 16, qbuf, W2s + 256, b2s + 16, bufA2);
  node_xform16_dual<<<xg1, xb, 0, stream>>>(bufB1, NT1, W2k + 256, b2k + 16, kbuf, W2v + 256, b2v + 16, vbuf);
  run_edge_passes(qbuf, kbuf, vbuf, W2e + 16, ea, ei1, ei2, alpha, mkey, sbuf, bufA2, NN2, stream);

  // ---------- layer 3 (only conv -> x1), params index 2 (W offset 512, b offset 32) ----------
  node_xform16_dual<<<xg1, xb, 0, stream>>>(bufA1, NT1, W2q + 512, b2q + 32, qbuf, W2s + 512, b2s + 32, bufB1);
  node_xform16_dual<<<xg2, xb, 0, stream>>>(bufA2, NT2, W2k + 512, b2k + 32, kbuf, W2v + 512, b2v + 32, vbuf);
  run_edge_passes(qbuf, kbuf, vbuf, W2e + 32, ea, ei2, ei1, alpha, mkey, sbuf, bufB1, NN1, stream);

  // ---------- final fc ----------
  fc_out<<<(NN1 + tb - 1) / tb, tb, 0, stream>>>(bufB1, fcw, fcb, out, NN1);
}